// AttentionalPoolingLayer_46110768889883
// MI455X (gfx1250) — compile-verified
//
#include <hip/hip_runtime.h>

typedef __attribute__((ext_vector_type(2))) float v2f;
typedef __attribute__((ext_vector_type(8))) float v8f;

#define B_    256
#define C_    256
#define N_    2048
#define HW_   16
#define NTILE 128          // 8 waves x 16 columns per workgroup
#define FPAD  17           // pad LDS rows to 17 dwords (bank-conflict free)

__global__ __launch_bounds__(256)
void attn_pool_fused(const float* __restrict__ features,
                     const float* __restrict__ weights,
                     float* __restrict__ out)
{
    // features_b as (c, hw) with padded rows; per-wave 16x16 attention tile
    __shared__ float featLDS[C_ * FPAD];            // ~17.0 KB
    __shared__ float attLDS[8 * HW_ * FPAD];        // ~8.7 KB

    const int b     = blockIdx.y;
    const int ntile = blockIdx.x;
    const int tid   = threadIdx.x;
    const int lane  = tid & 31;
    const int wave  = tid >> 5;
    const int half  = lane >> 4;     // 0: lanes 0-15, 1: lanes 16-31
    const int l15   = lane & 15;
    const int kk    = half * 2;      // K sub-offset for A/B fragments (16x16x4)

    // ---- stage 0: cache features_b (256 x 16) in LDS (reused by both GEMMs) ----
    const float* fb = features + (size_t)b * (C_ * HW_);
    for (int i = tid; i < C_ * HW_; i += 256) {
        featLDS[(i >> 4) * FPAD + (i & 15)] = fb[i];
    }
    __syncthreads();

    // ---- stage 1: scores(16 x 16 per wave) = feat^T (16x256) x W (256x16) ----
    // A[m=hw][k=c] from LDS ; B[k=c][n] streamed NT from HBM (used exactly once)
    const int    nglob = ntile * NTILE + wave * 16 + l15;
    const float* wptr  = weights + (size_t)b * C_ * N_ + (size_t)kk * N_ + nglob;

    v8f acc = {};
    #pragma unroll 8
    for (int k0 = 0; k0 < C_; k0 += 4) {
        v2f a, w;
        a.x = featLDS[(k0 + kk)     * FPAD + l15];
        a.y = featLDS[(k0 + kk + 1) * FPAD + l15];
        w.x = __builtin_nontemporal_load(wptr + (size_t)k0       * N_);
        w.y = __builtin_nontemporal_load(wptr + (size_t)(k0 + 1) * N_);
        acc = __builtin_amdgcn_wmma_f32_16x16x4_f32(false, a, false, w,
                                                    (short)0, acc, false, false);
    }

    // ---- softplus + const, column denominator (16 rows = this lane + lane^16), normalize ----
    float att[8];
    float s = 0.f;
    #pragma unroll
    for (int r = 0; r < 8; ++r) {
        float x  = acc[r];
        float sp = fmaxf(x, 0.f) + __logf(1.f + __expf(-fabsf(x)));  // stable softplus
        att[r]   = sp + 1e-4f;
        s += att[r];
    }
    float denom = s + __shfl_xor(s, 16, 32);   // column n sum over all 16 hw rows
    float inv   = __frcp_rn(denom) ;

    float* myAtt = attLDS + wave * (HW_ * FPAD);
    #pragma unroll
    for (int r = 0; r < 8; ++r) {             // D layout: row = r + 8*half, col = l15
        myAtt[(r + half * 8) * FPAD + l15] = att[r] * inv;
    }
    __syncthreads();

    // ---- stage 2: out_tile(256 x 16) = feat (256x16) x attmap (16x16) ----
    float* ob = out + (size_t)b * C_ * N_ + ntile * NTILE + wave * 16;
    #pragma unroll 1
    for (int mt = 0; mt < 16; ++mt) {
        const int m = mt * 16 + l15;          // A row = channel c
        v8f acc2 = {};
        #pragma unroll
        for (int k0 = 0; k0 < HW_; k0 += 4) {
            v2f a, bb;
            a.x  = featLDS[m * FPAD + (k0 + kk)];
            a.y  = featLDS[m * FPAD + (k0 + kk + 1)];
            bb.x = myAtt[(k0 + kk)     * FPAD + l15];
            bb.y = myAtt[(k0 + kk + 1) * FPAD + l15];
            acc2 = __builtin_amdgcn_wmma_f32_16x16x4_f32(false, a, false, bb,
                                                         (short)0, acc2, false, false);
        }
        #pragma unroll
        for (int r = 0; r < 8; ++r) {         // written once -> NT store, keep L2 for features
            __builtin_nontemporal_store(acc2[r],
                ob + (size_t)(mt * 16 + r + half * 8) * N_ + l15);
        }
    }
}

extern "C" void kernel_launch(void* const* d_in, const int* in_sizes, int n_in,
                              void* d_out, int out_size, void* d_ws, size_t ws_size,
                              hipStream_t stream) {
    const float* features = (const float*)d_in[0];   // (256, 256, 4, 4) fp32
    const float* weights  = (const float*)d_in[1];   // (256, 256, 2048) fp32
    float* out = (float*)d_out;                      // (256, 256, 2048) fp32

    dim3 grid(N_ / NTILE, B_);   // 16 n-tiles x 256 batches
    attn_pool_fused<<<grid, 256, 0, stream>>>(features, weights, out);
}